// Block_17901423690565
// MI455X (gfx1250) — compile-verified
//
#include <hip/hip_runtime.h>
#include <math.h>

#define A_DIM 128
#define V_DIM 64
#define R_DIM 32
#define NB    16
#define NN    4096
#define NROWS (NB * NN)
#define EPSV  1e-5f

typedef __attribute__((ext_vector_type(2))) float v2f;
typedef __attribute__((ext_vector_type(8))) float v8f;

// ---------------------------------------------------------------------------
// WMMA helpers: D = A(16x4 f32) x B(4x16 f32) + C(16x16 f32)
// A fragment: lane l holds rows m=l&15, K = kk*4 + ((l>>4)<<1) + {0,1}
// B fragment: lane l holds col o, K = kk*4 + ((l>>4)<<1) + {0,1} (packed layout)
// C/D: vgpr v -> m = v + ((l>>4)<<3), col = l&15 (+tile base)
// ---------------------------------------------------------------------------
__device__ __forceinline__ v8f wmma4(v2f a, v2f b, v8f c) {
  return __builtin_amdgcn_wmma_f32_16x16x4_f32(false, a, false, b, (short)0, c,
                                               false, false);
}

__device__ __forceinline__ v2f ldsA(const float* base, int stride, int lane, int kk) {
  int m  = lane & 15;
  int kb = kk * 4 + ((lane >> 4) << 1);
  const float* p = base + m * stride + kb;
  v2f r; r.x = p[0]; r.y = p[1];
  return r;
}

__device__ __forceinline__ v8f gemm_tile(const float* Abase, int astride,
                                         const float* __restrict__ Bpk, int KK,
                                         int lane, v8f c) {
#pragma unroll 8
  for (int kk = 0; kk < KK; ++kk) {
    v2f a = ldsA(Abase, astride, lane, kk);
    v2f b;
    const float* bp = Bpk + kk * 64 + lane * 2;
    b.x = bp[0]; b.y = bp[1];
    c = wmma4(a, b, c);
  }
  return c;
}

// ---------------------------------------------------------------------------
// Weight packer: src W[o][i][k] (row-major O,I,K) -> fragment-major packed:
// dst[ ((k*(O/16)+ot)*(I/4)+kk)*64 + lane*2 + j ] = W[ot*16+(lane&15)][kk*4+((lane>>4)<<1)+j][k]
// ---------------------------------------------------------------------------
__global__ void pack_w(const float* __restrict__ src, float* __restrict__ dst,
                       int O, int I, int K) {
  int idx = blockIdx.x * 256 + threadIdx.x;
  int total = O * I * K;
  if (idx >= total) return;
  int j    = idx & 1;
  int lane = (idx >> 1) & 31;
  int rest = idx >> 6;
  int I4 = I >> 2, O16 = O >> 4;
  int kk = rest % I4; rest /= I4;
  int ot = rest % O16;
  int k  = rest / O16;
  int o = ot * 16 + (lane & 15);
  int i = kk * 4 + ((lane >> 4) << 1) + j;
  dst[idx] = src[(o * I + i) * K + k];
}

__global__ void k_copy(const float* __restrict__ src, float* __restrict__ dst, int n) {
  int i = blockIdx.x * 256 + threadIdx.x;
  if (i < n) dst[i] = src[i];
}

// ---------------------------------------------------------------------------
// Edge embedding: per 16-edge tile, compute vecs/norms, h1 = lrelu(norms@w1+b1)
// (VALU, K=4), then ea = h1@w2^T + b2 via WMMA; ev = vlin(vecs, ee_lin_v).
// ---------------------------------------------------------------------------
__global__ void __launch_bounds__(256)
k_edge_embed(const float* __restrict__ pos0, const float* __restrict__ pos1,
             const float* __restrict__ ee_lin_v,
             const float* __restrict__ ee_w1, const float* __restrict__ ee_b1,
             const float* __restrict__ pk_ee2, const float* __restrict__ ee_b2,
             float* __restrict__ ea, float* __restrict__ ev) {
  __shared__ float vecs[16][4][3];
  __shared__ float norms[16][4];
  __shared__ float h1[16][A_DIM];
  int b  = blockIdx.x >> 8;
  int e0 = (blockIdx.x & 255) << 4;
  int tid = threadIdx.x;
  if (tid < 64) {
    int m = tid >> 2, j = tid & 3;
    int e = e0 + m;
    float vx = 0.f, vy = 0.f, vz = 0.f;
    if (e < NN - 1) {
      size_t r0 = ((size_t)b * NN + e) * 3;
      size_t r1 = r0 + 3;
      float ax, ay, az, bx, by, bz;
      if (j == 0)      { ax=pos0[r1]; ay=pos0[r1+1]; az=pos0[r1+2]; bx=pos0[r0]; by=pos0[r0+1]; bz=pos0[r0+2]; }
      else if (j == 1) { ax=pos1[r1]; ay=pos1[r1+1]; az=pos1[r1+2]; bx=pos1[r0]; by=pos1[r0+1]; bz=pos1[r0+2]; }
      else if (j == 2) { ax=pos1[r0]; ay=pos1[r0+1]; az=pos1[r0+2]; bx=pos0[r1]; by=pos0[r1+1]; bz=pos0[r1+2]; }
      else             { ax=pos1[r1]; ay=pos1[r1+1]; az=pos1[r1+2]; bx=pos0[r0]; by=pos0[r0+1]; bz=pos0[r0+2]; }
      vx = ax - bx; vy = ay - by; vz = az - bz;
    }
    vecs[m][j][0] = vx; vecs[m][j][1] = vy; vecs[m][j][2] = vz;
    norms[m][j] = sqrtf(vx * vx + vy * vy + vz * vz);
  }
  __syncthreads();
  for (int t = tid; t < 16 * A_DIM; t += 256) {
    int m = t >> 7, c = t & 127;
    float acc = ee_b1[c];
#pragma unroll
    for (int j = 0; j < 4; ++j) acc += norms[m][j] * ee_w1[c * 4 + j];
    h1[m][c] = acc >= 0.f ? acc : 0.2f * acc;
  }
  __syncthreads();
  int wave = tid >> 5, lane = tid & 31;
  {
    v8f acc = {};
    acc = gemm_tile(&h1[0][0], A_DIM, pk_ee2 + (size_t)wave * 32 * 64, 32, lane, acc);
    int col = (wave << 4) + (lane & 15);
    float bias = ee_b2[col];
    int mb = (lane >> 4) << 3;
#pragma unroll
    for (int v = 0; v < 8; ++v) {
      int e = e0 + mb + v;
      if (e < NN - 1) ea[((size_t)b * NN + e) * A_DIM + col] = acc[v] + bias;
    }
  }
  for (int t = tid; t < 16 * V_DIM * 3; t += 256) {
    int m = t / (V_DIM * 3);
    int rem = t - m * (V_DIM * 3);
    int ch = rem / 3, d = rem - ch * 3;
    float acc = 0.f;
#pragma unroll
    for (int j = 0; j < 4; ++j) acc += ee_lin_v[ch * 4 + j] * vecs[m][j][d];
    int e = e0 + m;
    if (e < NN - 1) ev[(((size_t)b * NN + e) * V_DIM + ch) * 3 + d] = acc;
  }
}

// ---------------------------------------------------------------------------
// y_a = conv7(x_a) + c0a_b + conv4(ea) + eeca_b + lrelu(nn*w1+b1)@w2^T + b2
// ---------------------------------------------------------------------------
__global__ void __launch_bounds__(256)
k_ya(const float* __restrict__ x_a, const float* __restrict__ ea,
     const float* __restrict__ pos0, const float* __restrict__ pos1,
     const float* __restrict__ pk_c0a, const float* __restrict__ pk_eeca,
     const float* __restrict__ pk_ne2,
     const float* __restrict__ c0a_b, const float* __restrict__ eeca_b,
     const float* __restrict__ ne_w1, const float* __restrict__ ne_b1,
     const float* __restrict__ ne_b2, float* __restrict__ ya_ws) {
  __shared__ float xs[22][A_DIM];
  __shared__ float es[19][A_DIM];
  __shared__ float hn[16][A_DIM];
  __shared__ float nnorm[16];
  int b  = blockIdx.x >> 8;
  int n0 = (blockIdx.x & 255) << 4;
  int tid = threadIdx.x;
  for (int t = tid; t < 22 * A_DIM; t += 256) {
    int r = t >> 7, c = t & 127;
    int n = n0 - 3 + r;
    xs[r][c] = (n >= 0 && n < NN) ? x_a[((size_t)b * NN + n) * A_DIM + c] : 0.f;
  }
  for (int t = tid; t < 19 * A_DIM; t += 256) {
    int r = t >> 7, c = t & 127;
    int e = n0 - 2 + r;
    es[r][c] = (e >= 0 && e < NN - 1) ? ea[((size_t)b * NN + e) * A_DIM + c] : 0.f;
  }
  if (tid < 16) {
    size_t base = ((size_t)b * NN + n0 + tid) * 3;
    float dx = pos1[base] - pos0[base];
    float dy = pos1[base + 1] - pos0[base + 1];
    float dz = pos1[base + 2] - pos0[base + 2];
    nnorm[tid] = sqrtf(dx * dx + dy * dy + dz * dz);
  }
  __syncthreads();
  for (int t = tid; t < 16 * A_DIM; t += 256) {
    int m = t >> 7, c = t & 127;
    float a = nnorm[m] * ne_w1[c] + ne_b1[c];
    hn[m][c] = a >= 0.f ? a : 0.2f * a;
  }
  __syncthreads();
  int ot = tid >> 5, lane = tid & 31;
  v8f acc = {};
  for (int k = 0; k < 7; ++k)
    acc = gemm_tile(&xs[k][0], A_DIM, pk_c0a + ((size_t)(k * 8 + ot)) * 32 * 64, 32, lane, acc);
  for (int k = 0; k < 4; ++k)
    acc = gemm_tile(&es[k][0], A_DIM, pk_eeca + ((size_t)(k * 8 + ot)) * 32 * 64, 32, lane, acc);
  acc = gemm_tile(&hn[0][0], A_DIM, pk_ne2 + (size_t)ot * 32 * 64, 32, lane, acc);
  int col = (ot << 4) + (lane & 15);
  float bias = c0a_b[col] + eeca_b[col] + ne_b2[col];
  int mb = (lane >> 4) << 3;
#pragma unroll
  for (int v = 0; v < 8; ++v)
    ya_ws[((size_t)b * NN + n0 + mb + v) * A_DIM + col] = acc[v] + bias;
}

// ---------------------------------------------------------------------------
// y_v = conv7(x_v) + conv4(ev) + rel * ne_lin_v  (per xyz component)
// 12 waves: wave -> (d = w>>2, ot = w&3)
// ---------------------------------------------------------------------------
__global__ void __launch_bounds__(384)
k_yv(const float* __restrict__ x_v, const float* __restrict__ ev,
     const float* __restrict__ pos0, const float* __restrict__ pos1,
     const float* __restrict__ pk_c0v, const float* __restrict__ pk_eecv,
     const float* __restrict__ ne_lin_v, float* __restrict__ yv_ws) {
  __shared__ float xv[22][3][V_DIM];
  __shared__ float evs[19][3][V_DIM];
  __shared__ float rel[16][3];
  int b  = blockIdx.x >> 8;
  int n0 = (blockIdx.x & 255) << 4;
  int tid = threadIdx.x;
  for (int t = tid; t < 22 * 3 * V_DIM; t += 384) {
    int r = t / 192, rem = t - r * 192;
    int d = rem >> 6, c = rem & 63;
    int n = n0 - 3 + r;
    xv[r][d][c] = (n >= 0 && n < NN) ? x_v[(((size_t)b * NN + n) * V_DIM + c) * 3 + d] : 0.f;
  }
  for (int t = tid; t < 19 * 3 * V_DIM; t += 384) {
    int r = t / 192, rem = t - r * 192;
    int d = rem >> 6, c = rem & 63;
    int e = n0 - 2 + r;
    evs[r][d][c] = (e >= 0 && e < NN - 1) ? ev[(((size_t)b * NN + e) * V_DIM + c) * 3 + d] : 0.f;
  }
  if (tid < 48) {
    int m = tid / 3, d = tid - m * 3;
    size_t base = ((size_t)b * NN + n0 + m) * 3 + d;
    rel[m][d] = pos1[base] - pos0[base];
  }
  __syncthreads();
  int w = tid >> 5, lane = tid & 31;
  int d = w >> 2, ot = w & 3;
  v8f acc = {};
  for (int k = 0; k < 7; ++k)
    acc = gemm_tile(&xv[k][d][0], 3 * V_DIM, pk_c0v + ((size_t)(k * 4 + ot)) * 16 * 64, 16, lane, acc);
  for (int k = 0; k < 4; ++k)
    acc = gemm_tile(&evs[k][d][0], 3 * V_DIM, pk_eecv + ((size_t)(k * 4 + ot)) * 16 * 64, 16, lane, acc);
  int col = (ot << 4) + (lane & 15);
  float nv = ne_lin_v[col];
  int mb = (lane >> 4) << 3;
#pragma unroll
  for (int v = 0; v < 8; ++v) {
    int m = mb + v;
    yv_ws[(((size_t)b * NN + n0 + m) * V_DIM + col) * 3 + d] = acc[v] + nv * rel[m][d];
  }
}

// ---------------------------------------------------------------------------
// LocalResidual: fully per-node, fused. Stages of independent 16x16 GEMM tasks
// distributed over 8 waves, barriers between dependency levels.
// ---------------------------------------------------------------------------
__global__ void __launch_bounds__(256)
k_local(float* __restrict__ ya_ws, float* __restrict__ yv_ws,
        const float* __restrict__ pk_la1, const float* __restrict__ la_b1,
        const float* __restrict__ pk_la2, const float* __restrict__ la_b2,
        const float* __restrict__ pk_la3, const float* __restrict__ la_b3,
        const float* __restrict__ pk_lv1, const float* __restrict__ pk_lv2,
        const float* __restrict__ pk_lv3,
        const float* __restrict__ pk_war, const float* __restrict__ av_bar,
        const float* __restrict__ pk_wq1, const float* __restrict__ pk_wq2,
        const float* __restrict__ pk_wra, const float* __restrict__ av_bra,
        const float* __restrict__ pk_wrv,
        const float* __restrict__ ga_g, const float* __restrict__ ga_b,
        const float* __restrict__ gv_g) {
  __shared__ float sya[16][A_DIM];
  __shared__ float t1[16][A_DIM];
  __shared__ float t2[16][A_DIM];
  __shared__ float sza[16][A_DIM];
  __shared__ float syv[16][3][V_DIM];
  __shared__ float u1[16][3][V_DIM];
  __shared__ float u2[16][3][V_DIM];
  __shared__ float szv[16][3][V_DIM];
  __shared__ float q1s[16][3][R_DIM];
  __shared__ float q2s[16][3][R_DIM];   // reused as s*q1 after E1
  __shared__ float dotq[16][R_DIM];
  __shared__ float ss[16][R_DIM];
  int b  = blockIdx.x >> 8;
  int n0 = (blockIdx.x & 255) << 4;
  size_t row0 = (size_t)b * NN + n0;
  int tid = threadIdx.x;
  int wave = tid >> 5, lane = tid & 31;
  for (int t = tid; t < 16 * A_DIM; t += 256) {
    int m = t >> 7, c = t & 127;
    sya[m][c] = ya_ws[(row0 + m) * A_DIM + c];
  }
  for (int t = tid; t < 16 * 3 * V_DIM; t += 256) {
    int m = t / 192, rem = t - m * 192;
    int d = rem >> 6, c = rem & 63;
    syv[m][d][c] = yv_ws[((row0 + m) * V_DIM + c) * 3 + d];
  }
  __syncthreads();
  // S1: t1(8) | q1(6) | q2(6) | s(2) | u1(12)  => 34 tasks
  for (int task = wave; task < 34; task += 8) {
    v8f acc = {};
    if (task < 8) {
      int ot = task;
      acc = gemm_tile(&sya[0][0], A_DIM, pk_la1 + (size_t)ot * 32 * 64, 32, lane, acc);
      int col = (ot << 4) + (lane & 15); float bias = la_b1[col]; int mb = (lane >> 4) << 3;
#pragma unroll
      for (int v = 0; v < 8; ++v) { float x = acc[v] + bias; t1[mb + v][col] = x >= 0.f ? x : 0.2f * x; }
    } else if (task < 14) {
      int idx = task - 8; int d = idx >> 1, ot = idx & 1;
      acc = gemm_tile(&syv[0][d][0], 192, pk_wq1 + (size_t)ot * 16 * 64, 16, lane, acc);
      int col = (ot << 4) + (lane & 15); int mb = (lane >> 4) << 3;
#pragma unroll
      for (int v = 0; v < 8; ++v) q1s[mb + v][d][col] = acc[v];
    } else if (task < 20) {
      int idx = task - 14; int d = idx >> 1, ot = idx & 1;
      acc = gemm_tile(&syv[0][d][0], 192, pk_wq2 + (size_t)ot * 16 * 64, 16, lane, acc);
      int col = (ot << 4) + (lane & 15); int mb = (lane >> 4) << 3;
#pragma unroll
      for (int v = 0; v < 8; ++v) q2s[mb + v][d][col] = acc[v];
    } else if (task < 22) {
      int ot = task - 20;
      acc = gemm_tile(&sya[0][0], A_DIM, pk_war + (size_t)ot * 32 * 64, 32, lane, acc);
      int col = (ot << 4) + (lane & 15); float bias = av_bar[col]; int mb = (lane >> 4) << 3;
#pragma unroll
      for (int v = 0; v < 8; ++v) ss[mb + v][col] = acc[v] + bias;
    } else {
      int idx = task - 22; int d = idx >> 2, ot = idx & 3;
      acc = gemm_tile(&syv[0][d][0], 192, pk_lv1 + (size_t)ot * 16 * 64, 16, lane, acc);
      int col = (ot << 4) + (lane & 15); int mb = (lane >> 4) << 3;
#pragma unroll
      for (int v = 0; v < 8; ++v) u1[mb + v][d][col] = acc[v];
    }
  }
  __syncthreads();
  // E1: vroots(u1) in place; dotq = sum_d q1*q2; q2s <- s*q1
  for (int t = tid; t < 16 * V_DIM; t += 256) {
    int m = t >> 6, c = t & 63;
    float a0 = u1[m][0][c], a1 = u1[m][1][c], a2 = u1[m][2][c];
    float scl = rsqrtf(sqrtf(1.f + a0 * a0 + a1 * a1 + a2 * a2));
    u1[m][0][c] = a0 * scl; u1[m][1][c] = a1 * scl; u1[m][2][c] = a2 * scl;
  }
  for (int t = tid; t < 16 * R_DIM; t += 256) {
    int m = t >> 5, r = t & 31;
    float d0 = 0.f; float sv = ss[m][r];
#pragma unroll
    for (int d = 0; d < 3; ++d) {
      float q1v = q1s[m][d][r];
      d0 += q1v * q2s[m][d][r];
      q2s[m][d][r] = sv * q1v;
    }
    dotq[m][r] = d0;
  }
  __syncthreads();
  // S2: t2(8) | u2(12) | p_a->sza(8) | p_v->szv(12)  => 40 tasks
  for (int task = wave; task < 40; task += 8) {
    v8f acc = {};
    if (task < 8) {
      int ot = task;
      acc = gemm_tile(&t1[0][0], A_DIM, pk_la2 + (size_t)ot * 32 * 64, 32, lane, acc);
      int col = (ot << 4) + (lane & 15); float bias = la_b2[col]; int mb = (lane >> 4) << 3;
#pragma unroll
      for (int v = 0; v < 8; ++v) { float x = acc[v] + bias; t2[mb + v][col] = x >= 0.f ? x : 0.2f * x; }
    } else if (task < 20) {
      int idx = task - 8; int d = idx >> 2, ot = idx & 3;
      acc = gemm_tile(&u1[0][d][0], 192, pk_lv2 + (size_t)ot * 16 * 64, 16, lane, acc);
      int col = (ot << 4) + (lane & 15); int mb = (lane >> 4) << 3;
#pragma unroll
      for (int v = 0; v < 8; ++v) u2[mb + v][d][col] = acc[v];
    } else if (task < 28) {
      int ot = task - 20;
      acc = gemm_tile(&dotq[0][0], R_DIM, pk_wra + (size_t)ot * 8 * 64, 8, lane, acc);
      int col = (ot << 4) + (lane & 15); float bias = av_bra[col]; int mb = (lane >> 4) << 3;
#pragma unroll
      for (int v = 0; v < 8; ++v) sza[mb + v][col] = acc[v] + bias;
    } else {
      int idx = task - 28; int d = idx >> 2, ot = idx & 3;
      acc = gemm_tile(&q2s[0][d][0], 96, pk_wrv + (size_t)ot * 8 * 64, 8, lane, acc);
      int col = (ot << 4) + (lane & 15); int mb = (lane >> 4) << 3;
#pragma unroll
      for (int v = 0; v < 8; ++v) szv[mb + v][d][col] = acc[v];
    }
  }
  __syncthreads();
  // E2: vroots(u2)
  for (int t = tid; t < 16 * V_DIM; t += 256) {
    int m = t >> 6, c = t & 63;
    float a0 = u2[m][0][c], a1 = u2[m][1][c], a2 = u2[m][2][c];
    float scl = rsqrtf(sqrtf(1.f + a0 * a0 + a1 * a1 + a2 * a2));
    u2[m][0][c] = a0 * scl; u2[m][1][c] = a1 * scl; u2[m][2][c] = a2 * scl;
  }
  __syncthreads();
  // S3: sza += t2@la3+b3 (8) | szv += u2@lv3 (12)  => 20 tasks
  for (int task = wave; task < 20; task += 8) {
    v8f acc = {};
    if (task < 8) {
      int ot = task;
      acc = gemm_tile(&t2[0][0], A_DIM, pk_la3 + (size_t)ot * 32 * 64, 32, lane, acc);
      int col = (ot << 4) + (lane & 15); float bias = la_b3[col]; int mb = (lane >> 4) << 3;
#pragma unroll
      for (int v = 0; v < 8; ++v) sza[mb + v][col] += acc[v] + bias;
    } else {
      int idx = task - 8; int d = idx >> 2, ot = idx & 3;
      acc = gemm_tile(&u2[0][d][0], 192, pk_lv3 + (size_t)ot * 16 * 64, 16, lane, acc);
      int col = (ot << 4) + (lane & 15); int mb = (lane >> 4) << 3;
#pragma unroll
      for (int v = 0; v < 8; ++v) szv[mb + v][d][col] += acc[v];
    }
  }
  __syncthreads();
  // E3: group norms + residual write-back (in place in ws)
  {
    int m = tid >> 4, g = tid & 15;
    float vals[8]; float mu = 0.f;
#pragma unroll
    for (int j = 0; j < 8; ++j) { vals[j] = sza[m][g * 8 + j]; mu += vals[j]; }
    mu *= 0.125f;
    float var = 0.f;
#pragma unroll
    for (int j = 0; j < 8; ++j) { float dl = vals[j] - mu; var += dl * dl; }
    var *= 0.125f;
    float inv = rsqrtf(var + EPSV);
#pragma unroll
    for (int j = 0; j < 8; ++j) {
      int c = g * 8 + j;
      float z = (vals[j] - mu) * inv * ga_g[c] + ga_b[c];
      ya_ws[(row0 + m) * A_DIM + c] = sya[m][c] + z;
    }
  }
  if (tid < 128) {
    int m = tid >> 3, g = tid & 7;
    float msq = 0.f;
#pragma unroll
    for (int j = 0; j < 8; ++j) {
      int c = g * 8 + j;
      float a0 = szv[m][0][c], a1 = szv[m][1][c], a2 = szv[m][2][c];
      msq += a0 * a0 + a1 * a1 + a2 * a2;
    }
    msq *= 0.125f;
    float inv = rsqrtf(msq + EPSV);
#pragma unroll
    for (int j = 0; j < 8; ++j) {
      int c = g * 8 + j;
      float gg = gv_g[c] * inv;
#pragma unroll
      for (int d = 0; d < 3; ++d)
        yv_ws[((row0 + m) * V_DIM + c) * 3 + d] = syv[m][d][c] + szv[m][d][c] * gg;
    }
  }
}

// ---------------------------------------------------------------------------
// Final: out_xa = x_a + gnorm(conv7(y_a)+b)
// ---------------------------------------------------------------------------
__global__ void __launch_bounds__(256)
k_out_a(const float* __restrict__ ya_ws, const float* __restrict__ x_a,
        const float* __restrict__ pk_c1a, const float* __restrict__ c1a_b,
        const float* __restrict__ ga_g, const float* __restrict__ ga_b,
        float* __restrict__ out_xa) {
  __shared__ float xs[22][A_DIM];
  __shared__ float cz[16][A_DIM];
  int b  = blockIdx.x >> 8;
  int n0 = (blockIdx.x & 255) << 4;
  size_t row0 = (size_t)b * NN + n0;
  int tid = threadIdx.x;
  for (int t = tid; t < 22 * A_DIM; t += 256) {
    int r = t >> 7, c = t & 127;
    int n = n0 - 3 + r;
    xs[r][c] = (n >= 0 && n < NN) ? ya_ws[((size_t)b * NN + n) * A_DIM + c] : 0.f;
  }
  __syncthreads();
  int ot = tid >> 5, lane = tid & 31;
  v8f acc = {};
  for (int k = 0; k < 7; ++k)
    acc = gemm_tile(&xs[k][0], A_DIM, pk_c1a + ((size_t)(k * 8 + ot)) * 32 * 64, 32, lane, acc);
  int col = (ot << 4) + (lane & 15);
  float bias = c1a_b[col];
  int mb = (lane >> 4) << 3;
#pragma unroll
  for (int v = 0; v < 8; ++v) cz[mb + v][col] = acc[v] + bias;
  __syncthreads();
  int m = tid >> 4, g = tid & 15;
  float vals[8]; float mu = 0.f;
#pragma unroll
  for (int j = 0; j < 8; ++j) { vals[j] = cz[m][g * 8 + j]; mu += vals[j]; }
  mu *= 0.125f;
  float var = 0.f;
#pragma unroll
  for (int j = 0; j < 8; ++j) { float dl = vals[j] - mu; var += dl * dl; }
  var *= 0.125f;
  float inv = rsqrtf(var + EPSV);
#pragma unroll
  for (int j = 0; j < 8; ++j) {
    int c = g * 8 + j;
    float z = (vals[j] - mu) * inv * ga_g[c] + ga_b[c];
    out_xa[(row0 + m) * A_DIM + c] = x_a[(row0 + m) * A_DIM + c] + z;
  }
}

// ---------------------------------------------------------------------------
// Final: out_xv = x_v + vec_gnorm(conv7(y_v))
// ---------------------------------------------------------------------------
__global__ void __launch_bounds__(384)
k_out_v(const float* __restrict__ yv_ws, const float* __restrict__ x_v,
        const float* __restrict__ pk_c1v, const float* __restrict__ gv_g,
        float* __restrict__ out_xv) {
  __shared__ float xv[22][3][V_DIM];
  __shared__ float cz[16][3][V_DIM];
  int b  = blockIdx.x >> 8;
  int n0 = (blockIdx.x & 255) << 4;
  size_t row0 = (size_t)b * NN + n0;
  int tid = threadIdx.x;
  for (int t = tid; t < 22 * 3 * V_DIM; t += 384) {
    int r = t / 192, rem = t - r * 192;
    int d = rem >> 6, c = rem & 63;
    int n = n0 - 3 + r;
    xv[r][d][c] = (n >= 0 && n < NN) ? yv_ws[(((size_t)b * NN + n) * V_DIM + c) * 3 + d] : 0.f;
  }
  __syncthreads();
  int w = tid >> 5, lane = tid & 31;
  int d = w >> 2, ot = w & 3;
  v8f acc = {};
  for (int k = 0; k < 7; ++k)
    acc = gemm_tile(&xv[k][d][0], 192, pk_c1v + ((size_t)(k * 4 + ot)) * 16 * 64, 16, lane, acc);
  int col = (ot << 4) + (lane & 15);
  int mb = (lane >> 4) << 3;
#pragma unroll
  for (int v = 0; v < 8; ++v) cz[mb + v][d][col] = acc[v];
  __syncthreads();
  if (tid < 128) {
    int m = tid >> 3, g = tid & 7;
    float msq = 0.f;
#pragma unroll
    for (int j = 0; j < 8; ++j) {
      int c = g * 8 + j;
      float a0 = cz[m][0][c], a1 = cz[m][1][c], a2 = cz[m][2][c];
      msq += a0 * a0 + a1 * a1 + a2 * a2;
    }
    msq *= 0.125f;
    float inv = rsqrtf(msq + EPSV);
#pragma unroll
    for (int j = 0; j < 8; ++j) {
      int c = g * 8 + j;
      float gg = gv_g[c] * inv;
#pragma unroll
      for (int dd = 0; dd < 3; ++dd)
        out_xv[((row0 + m) * V_DIM + c) * 3 + dd] =
            x_v[((row0 + m) * V_DIM + c) * 3 + dd] + cz[m][dd][c] * gg;
    }
  }
}

// ---------------------------------------------------------------------------
extern "C" void kernel_launch(void* const* d_in, const int* in_sizes, int n_in,
                              void* d_out, int out_size, void* d_ws, size_t ws_size,
                              hipStream_t stream) {
  (void)in_sizes; (void)n_in; (void)out_size; (void)ws_size;
  const float* pos0 = (const float*)d_in[0];
  const float* pos1 = (const float*)d_in[1];
  const float* x_a  = (const float*)d_in[2];
  const float* x_v  = (const float*)d_in[3];
  const float* ee_lin_v     = (const float*)d_in[4];
  const float* ee_w1        = (const float*)d_in[5];
  const float* ee_b1        = (const float*)d_in[6];
  const float* ee_w2        = (const float*)d_in[7];
  const float* ee_b2        = (const float*)d_in[8];
  const float* ee_conv_a_w  = (const float*)d_in[9];
  const float* ee_conv_a_b  = (const float*)d_in[10];
  const float* ee_conv_v_w  = (const float*)d_in[11];
  const float* ne_lin_v     = (const float*)d_in[12];
  const float* ne_w1        = (const float*)d_in[13];
  const float* ne_b1        = (const float*)d_in[14];
  const float* ne_w2        = (const float*)d_in[15];
  const float* ne_b2        = (const float*)d_in[16];
  const float* c0a_w        = (const float*)d_in[17];
  const float* c0a_b        = (const float*)d_in[18];
  const float* c0v_w        = (const float*)d_in[19];
  const float* la_w1        = (const float*)d_in[20];
  const float* la_b1        = (const float*)d_in[21];
  const float* la_w2        = (const float*)d_in[22];
  const float* la_b2        = (const float*)d_in[23];
  const float* la_w3        = (const float*)d_in[24];
  const float* la_b3        = (const float*)d_in[25];
  const float* lv_w1        = (const float*)d_in[26];
  const float* lv_w2        = (const float*)d_in[27];
  const float* lv_w3        = (const float*)d_in[28];
  const float* av_war       = (const float*)d_in[29];
  const float* av_bar       = (const float*)d_in[30];
  const float* av_wq1       = (const float*)d_in[31];
  const float* av_wq2       = (const float*)d_in[32];
  const float* av_wra       = (const float*)d_in[33];
  const float* av_bra       = (const float*)d_in[34];
  const float* av_wrv       = (const float*)d_in[35];
  const float* lr_ga_g      = (const float*)d_in[36];
  const float* lr_ga_b      = (const float*)d_in[37];
  const float* lr_gv_g      = (const float*)d_in[38];
  const float* c1a_w        = (const float*)d_in[39];
  const float* c1a_b        = (const float*)d_in[40];
  const float* c1v_w        = (const float*)d_in[41];
  const float* blk_ga_g     = (const float*)d_in[42];
  const float* blk_ga_b     = (const float*)d_in[43];
  const float* blk_gv_g     = (const float*)d_in[44];

  float* ws = (float*)d_ws;
  float* ea = ws;
  float* ev = ea + (size_t)NROWS * A_DIM;
  float* ya = ev + (size_t)NROWS * V_DIM * 3;
  float* yv = ya + (size_t)NROWS * A_DIM;
  float* pk = yv + (size_t)NROWS * V_DIM * 3;

  size_t off = 0;
  auto PK = [&](const float* src, int O, int I, int K) {
    float* dst = pk + off;
    off += (size_t)O * I * K;
    int total = O * I * K;
    pack_w<<<dim3((total + 255) / 256), dim3(256), 0, stream>>>(src, dst, O, I, K);
    return dst;
  };
  float* pk_ee2  = PK(ee_w2, 128, 128, 1);
  float* pk_ne2  = PK(ne_w2, 128, 128, 1);
  float* pk_c0a  = PK(c0a_w, 128, 128, 7);
  float* pk_eeca = PK(ee_conv_a_w, 128, 128, 4);
  float* pk_la1  = PK(la_w1, 128, 128, 1);
  float* pk_la2  = PK(la_w2, 128, 128, 1);
  float* pk_la3  = PK(la_w3, 128, 128, 1);
  float* pk_lv1  = PK(lv_w1, 64, 64, 1);
  float* pk_lv2  = PK(lv_w2, 64, 64, 1);
  float* pk_lv3  = PK(lv_w3, 64, 64, 1);
  float* pk_war  = PK(av_war, 32, 128, 1);
  float* pk_wra  = PK(av_wra, 128, 32, 1);
  float* pk_wq1  = PK(av_wq1, 32, 64, 1);
  float* pk_wq2  = PK(av_wq2, 32, 64, 1);
  float* pk_wrv  = PK(av_wrv, 64, 32, 1);
  float* pk_c0v  = PK(c0v_w, 64, 64, 7);
  float* pk_eecv = PK(ee_conv_v_w, 64, 64, 4);
  float* pk_c1a  = PK(c1a_w, 128, 128, 7);
  float* pk_c1v  = PK(c1v_w, 64, 64, 7);

  float* out = (float*)d_out;
  float* out_p0 = out;
  float* out_p1 = out + (size_t)NROWS * 3;
  float* out_xa = out + (size_t)NROWS * 6;
  float* out_xv = out_xa + (size_t)NROWS * A_DIM;

  int npos = NROWS * 3;
  k_copy<<<dim3((npos + 255) / 256), dim3(256), 0, stream>>>(pos0, out_p0, npos);
  k_copy<<<dim3((npos + 255) / 256), dim3(256), 0, stream>>>(pos1, out_p1, npos);

  k_edge_embed<<<dim3(NB * 256), dim3(256), 0, stream>>>(
      pos0, pos1, ee_lin_v, ee_w1, ee_b1, pk_ee2, ee_b2, ea, ev);
  k_ya<<<dim3(NB * 256), dim3(256), 0, stream>>>(
      x_a, ea, pos0, pos1, pk_c0a, pk_eeca, pk_ne2, c0a_b, ee_conv_a_b,
      ne_w1, ne_b1, ne_b2, ya);
  k_yv<<<dim3(NB * 256), dim3(384), 0, stream>>>(
      x_v, ev, pos0, pos1, pk_c0v, pk_eecv, ne_lin_v, yv);
  k_local<<<dim3(NB * 256), dim3(256), 0, stream>>>(
      ya, yv, pk_la1, la_b1, pk_la2, la_b2, pk_la3, la_b3,
      pk_lv1, pk_lv2, pk_lv3, pk_war, av_bar, pk_wq1, pk_wq2,
      pk_wra, av_bra, pk_wrv, lr_ga_g, lr_ga_b, lr_gv_g);
  k_out_a<<<dim3(NB * 256), dim3(256), 0, stream>>>(
      ya, x_a, pk_c1a, c1a_b, blk_ga_g, blk_ga_b, out_xa);
  k_out_v<<<dim3(NB * 256), dim3(384), 0, stream>>>(
      yv, x_v, pk_c1v, blk_gv_g, out_xv);
}